// VectorQuantizer_45586782880016
// MI455X (gfx1250) — compile-verified
//
#include <hip/hip_runtime.h>

typedef __attribute__((ext_vector_type(2))) float v2f;
typedef __attribute__((ext_vector_type(4))) float v4f;
typedef __attribute__((ext_vector_type(8))) float v8f;

#define N_E        1024
#define EDIM       64
#define ROWP       68      // padded LDS row stride (floats): breaks 64-bank stride conflicts
#define TOK_PER_WG 128
#define N_TOKENS   131072

__global__ __launch_bounds__(256) void vq_wmma_kernel(const float* __restrict__ z,
                                                      const float* __restrict__ emb,
                                                      float* __restrict__ zq,
                                                      float* __restrict__ out_idx)
{
    extern __shared__ float lds[];                 // [N_E*ROWP] codebook (+norm at col 64), then idx scratch
    float* ldse = lds;
    int*   idx_scratch = (int*)(lds + N_E * ROWP); // 8 waves * 16 tokens

    const int tid  = threadIdx.x;
    const int lane = tid & 31;
    const int wave = tid >> 5;

    // ---- 1) cooperative codebook load: global -> LDS (padded rows), float4 coalesced ----
    #pragma unroll 4
    for (int i = 0; i < 64; ++i) {
        int v = i * 256 + tid;        // float4 index 0..16383
        int f = v * 4;                // flat float index
        int n = f >> 6;               // code row
        int k = f & 63;               // col (multiple of 4)
        v4f val = *(const v4f*)(emb + f);
        *(v4f*)(ldse + n * ROWP + k) = val;
    }
    __syncthreads();

    // ---- 2) per-code squared norms into the pad slot ----
    for (int n = tid; n < N_E; n += 256) {
        float s = 0.f;
        #pragma unroll
        for (int k = 0; k < EDIM; ++k) { float e = ldse[n * ROWP + k]; s += e * e; }
        ldse[n * ROWP + 64] = s;
    }
    __syncthreads();

    // ---- 3) per-wave 16-token tile: A fragments resident in VGPRs ----
    const int tokbase = blockIdx.x * TOK_PER_WG + wave * 16;
    const int m  = lane & 15;         // token row within tile / code col within tile
    const int hi = lane >> 4;         // K-pair select per 16x16x4 f32 A/B layout

    v2f a[16];
    {
        const float* zrow = z + (size_t)(tokbase + m) * EDIM + hi * 2;
        #pragma unroll
        for (int kk = 0; kk < 16; ++kk)
            a[kk] = *(const v2f*)(zrow + kk * 4);
    }

    float minv[8];
    int   mini[8];
    #pragma unroll
    for (int r = 0; r < 8; ++r) { minv[r] = 3.4e38f; mini[r] = 0; }

    // ---- 4) sweep 64 code tiles: two independent 8-deep WMMA chains per tile ----
    for (int nb = 0; nb < 64; ++nb) {
        const float* bbase = ldse + (nb * 16 + m) * ROWP + hi * 2;
        v8f acc0 = {};
        v8f acc1 = {};
        #pragma unroll
        for (int kk = 0; kk < 8; ++kk) {
            v2f b0 = *(const v2f*)(bbase + (2 * kk) * 4);
            v2f b1 = *(const v2f*)(bbase + (2 * kk + 1) * 4);
            acc0 = __builtin_amdgcn_wmma_f32_16x16x4_f32(
                false, a[2 * kk],     false, b0, (short)0, acc0, false, false);
            acc1 = __builtin_amdgcn_wmma_f32_16x16x4_f32(
                false, a[2 * kk + 1], false, b1, (short)0, acc1, false, false);
        }
        float en   = ldse[(nb * 16 + m) * ROWP + 64];
        int   code = nb * 16 + m;     // this lane's code column, fixed for all 8 rows
        #pragma unroll
        for (int r = 0; r < 8; ++r) {
            float dot = acc0[r] + acc1[r];
            float d   = fmaf(-2.0f, dot, en);  // ||e||^2 - 2 z.e  (||z||^2 is row-constant)
            if (d < minv[r]) { minv[r] = d; mini[r] = code; }
        }
    }

    // ---- 5) argmin across the 16 lanes of each half-wave (wave32 shuffles) ----
    #pragma unroll
    for (int r = 0; r < 8; ++r) {
        float v = minv[r]; int ix = mini[r];
        #pragma unroll
        for (int off = 8; off >= 1; off >>= 1) {
            float ov = __shfl_xor(v, off, 32);
            int   oi = __shfl_xor(ix, off, 32);
            if (ov < v || (ov == v && oi < ix)) { v = ov; ix = oi; }  // tie -> lowest index
        }
        minv[r] = v; mini[r] = ix;
    }

    // ---- 6) lane n==0 of each half holds 8 token results; emit idx + stash for gather ----
    if (m == 0) {
        #pragma unroll
        for (int r = 0; r < 8; ++r) {
            int t = r + hi * 8;                   // token row 0..15 in this wave's tile
            idx_scratch[wave * 16 + t] = mini[r];
            out_idx[tokbase + t] = (float)mini[r];
        }
    }
    __syncthreads();

    // ---- 7) gather z_q rows from LDS codebook, coalesced float4 stores ----
    {
        int t    = lane >> 1;                     // token 0..15
        int j0   = (lane & 1) * 32;               // half-row
        int code = idx_scratch[wave * 16 + t];
        const float* src = ldse + code * ROWP + j0;
        float*       dst = zq + (size_t)(tokbase + t) * EDIM + j0;
        #pragma unroll
        for (int q = 0; q < 8; ++q)
            *(v4f*)(dst + q * 4) = *(const v4f*)(src + q * 4);
    }
}

extern "C" void kernel_launch(void* const* d_in, const int* in_sizes, int n_in,
                              void* d_out, int out_size, void* d_ws, size_t ws_size,
                              hipStream_t stream) {
    const float* z   = (const float*)d_in[0];   // [8,16,32,32,64] fp32
    const float* emb = (const float*)d_in[1];   // [1024,64] fp32
    float* zq   = (float*)d_out;                          // z_q: 131072*64 floats
    float* oidx = (float*)d_out + (size_t)N_TOKENS * EDIM; // idx (as float), 131072

    size_t shmem = (size_t)N_E * ROWP * sizeof(float) + 8 * 16 * sizeof(int);
    vq_wmma_kernel<<<N_TOKENS / TOK_PER_WG, 256, shmem, stream>>>(z, emb, zq, oidx);
}